// MinimalConvWTA_LIF_17059610100009
// MI455X (gfx1250) — compile-verified
//
#include <hip/hip_runtime.h>
#include <hip/hip_bf16.h>

typedef float v2f __attribute__((ext_vector_type(2)));
typedef float v8f __attribute__((ext_vector_type(8)));

#define ALPHA 0.95f
#define THETA 0.05f

// ----- CDNA5 async LDS<->global helpers (gfx1250 ISA §15.18 / §10) ---------
__device__ __forceinline__ void async_load_b64(unsigned lds_addr, unsigned goff,
                                               const void* sbase) {
    // GLOBAL_LOAD_ASYNC_TO_LDS_B64: vdst = LDS byte address, vaddr = u32
    // offset, saddr = 64-bit base. Tracked on ASYNCcnt.
    asm volatile("global_load_async_to_lds_b64 %0, %1, %2"
                 :: "v"(lds_addr), "v"(goff), "s"(sbase) : "memory");
}
__device__ __forceinline__ void async_store_b64(unsigned goff, unsigned lds_addr,
                                                void* sbase) {
    asm volatile("global_store_async_from_lds_b64 %0, %1, %2"
                 :: "v"(goff), "v"(lds_addr), "s"(sbase) : "memory");
}
__device__ __forceinline__ void wait_async0() {
    asm volatile("s_wait_asynccnt 0x0" ::: "memory");
}
__device__ __forceinline__ unsigned lds_addr_of(const void* p) {
    // Generic pointer to LDS: low 32 bits are the LDS byte offset (aperture rule).
    return (unsigned)(unsigned long long)p;
}

// ---------------------------------------------------------------------------
// Conv: u[b,k,t] = sum_{d=0}^{31} Wm[k][d] * x[b,t-d]   (delay form)
// One wave computes a 64-wide t-tile for one row: 4 sub-tiles x 8 chained
// V_WMMA_F32_16X16X4_F32 (A = weight slab 16x4, B = x window 4x16).
// ---------------------------------------------------------------------------
__global__ __launch_bounds__(256) void conv_wmma_kernel(
    const float* __restrict__ x,
    const float* __restrict__ w8,
    const float* __restrict__ w16,
    const float* __restrict__ w32,
    float* __restrict__ u, int B, int T)
{
    __shared__ float Wm[16 * 32];   // [channel m][delay d], zero for m >= 3
    __shared__ float xs[8][96];     // per-wave x window: t0-32 .. t0+63

    const int tid  = threadIdx.x;
    const int wave = tid >> 5;
    const int lane = tid & 31;

    for (int f = tid; f < 16 * 32; f += 256) {
        int m = f >> 5, d = f & 31;
        float val = 0.0f;
        if      (m == 0) { if (d < 8)  val = w8[7 - d];   }
        else if (m == 1) { if (d < 16) val = w16[15 - d]; }
        else if (m == 2) { if (d < 32) val = w32[31 - d]; }
        Wm[f] = val;
    }

    const long gw          = (long)blockIdx.x * 8 + wave;
    const int  tilesPerRow = T >> 6;               // 64 outputs per wave
    const int  row         = (int)(gw / tilesPerRow);
    const int  t0          = (int)(gw % tilesPerRow) << 6;
    const float* xr        = x + (long)row * T;

    {   // stage x[t0-32 .. t0+63]; only the causal left edge needs padding
        int g = t0 - 32 + lane;
        xs[wave][lane]      = (g >= 0) ? xr[g] : 0.0f;
        xs[wave][lane + 32] = xr[t0 + lane];
        xs[wave][lane + 64] = xr[t0 + 32 + lane];
    }
    __syncthreads();

    const int mrow = lane & 15;   // A: channel row
    const int ncol = lane & 15;   // B/C: time column within sub-tile
    const int kh   = lane >> 4;   // K-half per ISA 16x4 f32 layout
    const float* xw = &xs[wave][32];

    v8f c[4] = {};
    #pragma unroll
    for (int i = 0; i < 8; ++i) {
        const int d0 = 4 * i + 2 * kh;
        v2f a;
        a.x = Wm[mrow * 32 + d0];       // A[m][k], k = comp + 2*kh
        a.y = Wm[mrow * 32 + d0 + 1];
        #pragma unroll
        for (int j = 0; j < 4; ++j) {   // 4 sub-tiles share the A fragment
            v2f b;
            b.x = xw[16 * j + ncol - d0];        // B[k][n] = x[t0+16j+n-(4i+k)]
            b.y = xw[16 * j + ncol - d0 - 1];
            c[j] = __builtin_amdgcn_wmma_f32_16x16x4_f32(
                       false, a, false, b, (short)0, c[j], false, false);
        }
    }

    // C layout: VGPR r, lanes 0-15 => C[M=r][N=lane]; channels are r=0..2.
    if (lane < 16) {
        long base = ((long)row * 3) * T + t0 + lane;
        #pragma unroll
        for (int j = 0; j < 4; ++j) {
            u[base + 16 * j]          = c[j][0];
            u[base + T + 16 * j]      = c[j][1];
            u[base + 2L * T + 16 * j] = c[j][2];
        }
    }
}

// ---------------------------------------------------------------------------
// LIF + WTA scan, T chunked with warmup (alpha^512 ~ 4e-12). 128 lanes own
// 128 rows; u/s tiles move via GLOBAL_*_ASYNC_*_LDS_B64 (ASYNCcnt), with a
// 50-float LDS row pitch: 8B-aligned rows, and gcd(50,64)=2 => 32 distinct
// banks across the 32 lanes during per-lane streaming.
// ---------------------------------------------------------------------------
#define ROWS 128
#define TT   16
#define CH   512
#define WARM 512
#define PITCH 50

__global__ __launch_bounds__(ROWS) void lif_scan_kernel(
    const float* __restrict__ u, float* __restrict__ s, int B, int T)
{
    __shared__ float uls[ROWS * PITCH];
    __shared__ float sls[ROWS * PITCH];

    const int tid       = threadIdx.x;
    const int nRowGrp   = B / ROWS;
    const int rg        = blockIdx.x % nRowGrp;
    const int ch        = blockIdx.x / nRowGrp;
    const int row0      = rg * ROWS;
    const int tOutStart = ch * CH;
    const int tOutEnd   = tOutStart + CH;
    int tW = tOutStart - WARM; if (tW < 0) tW = 0;

    const unsigned uls_base = lds_addr_of(&uls[0]);
    const unsigned sls_base = lds_addr_of(&sls[0]);

    float v0 = 0.0f, v1 = 0.0f, v2 = 0.0f;

    for (int tb = tW; tb < tOutEnd; tb += TT) {
        __syncthreads();   // previous tile fully consumed (incl. async stores)

        // Async cooperative load of u[row0..+127][0..2][tb..tb+15] (b64 units).
        #pragma unroll 4
        for (int it = 0; it < 3 * TT / 2; ++it) {     // 24 b64 per thread
            int f    = tid + ROWS * it;
            int tt2  = f & 7;                          // b64 index in segment
            int segk = f >> 3;
            int k    = segk % 3;
            int lr   = segk / 3;
            unsigned goff  = (unsigned)(((((long)(row0 + lr) * 3 + k) * T) + tb) * 4)
                           + (unsigned)(tt2 * 8);
            unsigned laddr = uls_base + (unsigned)((lr * PITCH + k * TT) * 4 + tt2 * 8);
            async_load_b64(laddr, goff, u);
        }
        __builtin_prefetch(&u[((long)(row0 + tid) * 3) * T + tb + TT], 0, 0);
        wait_async0();
        __syncthreads();

        const bool emit = (tb >= tOutStart);   // uniform across block
        const float* myu = &uls[tid * PITCH];
        float*       mys = &sls[tid * PITCH];

        #pragma unroll
        for (int tt = 0; tt < TT; ++tt) {
            v0 = ALPHA * v0 + myu[tt];
            v1 = ALPHA * v1 + myu[TT + tt];
            v2 = ALPHA * v2 + myu[2 * TT + tt];
            float m0 = v0 - THETA, m1 = v1 - THETA, m2 = v2 - THETA;
            int   kstar = 0; float mm = m0;            // first-occurrence ties
            if (m1 > mm) { mm = m1; kstar = 1; }
            if (m2 > mm) { mm = m2; kstar = 2; }
            float spike = (mm >= 0.0f) ? 1.0f : 0.0f;
            float s0 = (kstar == 0) ? spike : 0.0f;
            float s1 = (kstar == 1) ? spike : 0.0f;
            float s2 = (kstar == 2) ? spike : 0.0f;
            v0 -= s0 * THETA; v1 -= s1 * THETA; v2 -= s2 * THETA;
            if (emit) { mys[tt] = s0; mys[TT + tt] = s1; mys[2 * TT + tt] = s2; }
        }

        if (emit) {
            __syncthreads();   // sls tile complete across all lanes
            #pragma unroll 4
            for (int it = 0; it < 3 * TT / 2; ++it) {
                int f    = tid + ROWS * it;
                int tt2  = f & 7;
                int segk = f >> 3;
                int k    = segk % 3;
                int lr   = segk / 3;
                unsigned goff  = (unsigned)(((((long)(row0 + lr) * 3 + k) * T) + tb) * 4)
                               + (unsigned)(tt2 * 8);
                unsigned laddr = sls_base + (unsigned)((lr * PITCH + k * TT) * 4 + tt2 * 8);
                async_store_b64(goff, laddr, s);
            }
            wait_async0();     // sls reusable after next loop-top barrier
        }
    }
}

extern "C" void kernel_launch(void* const* d_in, const int* in_sizes, int n_in,
                              void* d_out, int out_size, void* d_ws, size_t ws_size,
                              hipStream_t stream) {
    (void)n_in; (void)out_size; (void)d_ws; (void)ws_size;
    const float* x   = (const float*)d_in[0];
    const float* w8  = (const float*)d_in[1];
    const float* w16 = (const float*)d_in[2];
    const float* w32 = (const float*)d_in[3];

    const int T = 4096;
    const int B = in_sizes[0] / T;     // x is [B,1,T] flat

    float* u = (float*)d_out;                 // output 0: u [B,3,T]
    float* s = u + (long)B * 3 * T;           // output 1: s [B,3,T]

    // Conv: one wave per (row, 64-t tile); 8 waves per block.
    long waves = (long)B * (T / 64);
    conv_wmma_kernel<<<dim3((unsigned)(waves / 8)), 256, 0, stream>>>(
        x, w8, w16, w32, u, B, T);

    // Scan: (B/ROWS) row groups x (T/CH) chunks.
    lif_scan_kernel<<<dim3((unsigned)((B / ROWS) * (T / CH))), ROWS, 0, stream>>>(
        u, s, B, T);
}